// LSA_40389872452054
// MI455X (gfx1250) — compile-verified
//
#include <hip/hip_runtime.h>
#include <hip/hip_bf16.h>
#include <math.h>

#define BB 64
#define TT 2048
#define RNN 1024
#define EMB 512
#define AD 128
#define NF 32
#define KS 31
#define PAD 15
#define TTILE 128
#define NSEG 4

typedef __attribute__((ext_vector_type(2))) float v2f;
typedef __attribute__((ext_vector_type(8))) float v8f;

// ---------------- kernel 1: processed query = h @ Wq^T  (64 x 128, K=1024)
__global__ void lsa_pq(const float* __restrict__ h, const float* __restrict__ wq,
                       float* __restrict__ pq) {
    __shared__ float hl[RNN];
    int b = blockIdx.x;
    int d = threadIdx.x;               // 128 threads
    for (int i = threadIdx.x; i < RNN; i += 128) hl[i] = h[b * RNN + i];
    __syncthreads();
    float s = 0.f;
    const float* wr = wq + (long)d * RNN;
    #pragma unroll 8
    for (int k = 0; k < RNN; ++k) s = fmaf(hl[k], wr[k], s);
    pq[b * AD + d] = s;
}

// ---------------- kernel 2: fused conv -> WMMA dense -> tanh energy -> alignment
__global__ void lsa_energy(const float* __restrict__ aw_cat,   // (B,2,T)
                           const float* __restrict__ conv_w,   // (32,2,31)
                           const float* __restrict__ wdense,   // (128,32)
                           const float* __restrict__ pq,       // (B,128)
                           const float* __restrict__ pmem,     // (B,T,128)
                           const float* __restrict__ wv,       // (128)
                           float* __restrict__ align_out)      // (B,T)
{
    __shared__ float aw_lds[2][TTILE + KS - 1];     // padded input window
    __shared__ float cw_lds[NF * 2 * KS];           // [f*62 + c*31 + k]
    __shared__ float loc_lds[TTILE][NF + 1];        // pad 33 -> conflict free
    __shared__ float wd_lds[NF][AD + 1];            // transposed W_dense, pad 129
    __shared__ float pq_lds[AD];
    __shared__ float wv_lds[AD];

    const int b   = blockIdx.x;
    const int t0  = blockIdx.y * TTILE;
    const int tid = threadIdx.x;                    // 256 threads = 8 waves

    // ---- stage LDS
    for (int i = tid; i < 2 * (TTILE + KS - 1); i += 256) {
        int c = i / (TTILE + KS - 1);
        int j = i - c * (TTILE + KS - 1);
        int tg = t0 + j - PAD;
        aw_lds[c][j] = (tg >= 0 && tg < TT) ? aw_cat[((long)b * 2 + c) * TT + tg] : 0.f;
    }
    for (int i = tid; i < NF * 2 * KS; i += 256) cw_lds[i] = conv_w[i];
    for (int i = tid; i < AD * NF; i += 256) {      // wd_lds[f][d] = wdense[d][f]
        int d = i >> 5, f = i & 31;
        wd_lds[f][d] = wdense[d * NF + f];
    }
    if (tid < AD) { pq_lds[tid] = pq[b * AD + tid]; wv_lds[tid] = wv[tid]; }
    __syncthreads();

    // ---- location conv: loc[t][f], 16 outputs/thread, broadcast aw reads
    #pragma unroll
    for (int i = 0; i < 16; ++i) {
        int idx = i * 256 + tid;                    // lanes share t, f = lane
        int t = idx >> 5, f = idx & 31;
        float s = 0.f;
        const float* cw0 = &cw_lds[f * 2 * KS];
        #pragma unroll
        for (int k = 0; k < KS; ++k) {
            s = fmaf(aw_lds[0][t + k], cw0[k], s);
            s = fmaf(aw_lds[1][t + k], cw0[KS + k], s);
        }
        loc_lds[t][f] = s;
    }
    __syncthreads();

    // ---- WMMA f32 16x16x4: C[t][d] = sum_f loc[t][f] * W_dense[d][f]
    const int wave = tid >> 5;
    const int lane = tid & 31;
    const int mrow = lane & 15;                     // A row / B,C column
    const int koff = (lane < 16) ? 0 : 2;           // ISA 32-bit A/B layout
    const int trow_base = wave * 16;                // each wave owns 16 t-rows

    float align_acc[8];
    #pragma unroll
    for (int r = 0; r < 8; ++r) align_acc[r] = 0.f;

    for (int nt = 0; nt < 8; ++nt) {                // 8 d-tiles of 16
        v8f c = {};
        #pragma unroll
        for (int kb = 0; kb < NF; kb += 4) {        // 8 chained wmmas
            v2f a, bm;
            a.x  = loc_lds[trow_base + mrow][kb + koff];
            a.y  = loc_lds[trow_base + mrow][kb + koff + 1];
            bm.x = wd_lds[kb + koff][nt * 16 + mrow];
            bm.y = wd_lds[kb + koff + 1][nt * 16 + mrow];
            c = __builtin_amdgcn_wmma_f32_16x16x4_f32(
                    false, a, false, bm, (short)0, c, false, false);
        }
        // epilogue: c[r] -> t = trow_base + (lane<16 ? r : r+8), d = nt*16+mrow
        int d = nt * 16 + mrow;
        float qv  = pq_lds[d];
        float wvv = wv_lds[d];
        #pragma unroll
        for (int r = 0; r < 8; ++r) {
            int t = trow_base + ((lane < 16) ? r : r + 8);
            float e = c[r] + qv + pmem[((long)b * TT + t0 + t) * AD + d];
            align_acc[r] += tanhf(e) * wvv;
        }
    }

    // ---- reduce over the 16 lanes holding the 16 d's of each half
    #pragma unroll
    for (int r = 0; r < 8; ++r) {
        float v = align_acc[r];
        v += __shfl_xor(v, 1, 32);
        v += __shfl_xor(v, 2, 32);
        v += __shfl_xor(v, 4, 32);
        v += __shfl_xor(v, 8, 32);
        if ((lane & 15) == 0) {
            int t = trow_base + ((lane < 16) ? r : r + 8);
            align_out[(long)b * TT + t0 + t] = v;
        }
    }
}

// ---------------- kernel 3: softmax over T per b; writes weights to d_out
__global__ void lsa_softmax(const float* __restrict__ align_in,
                            const unsigned char* __restrict__ mask,
                            float* __restrict__ weights)        // (B,T)
{
    __shared__ float red[256];
    int b = blockIdx.x, tid = threadIdx.x;
    const float* ab = align_in + (long)b * TT;
    const unsigned char* mb = mask + (long)b * TT;

    float m = -INFINITY;
    for (int t = tid; t < TT; t += 256) {
        float a = mb[t] ? -INFINITY : ab[t];
        m = fmaxf(m, a);
    }
    red[tid] = m; __syncthreads();
    for (int s = 128; s > 0; s >>= 1) {
        if (tid < s) red[tid] = fmaxf(red[tid], red[tid + s]);
        __syncthreads();
    }
    float mx = red[0]; __syncthreads();

    float sum = 0.f;
    for (int t = tid; t < TT; t += 256) {
        float a = mb[t] ? -INFINITY : ab[t];
        sum += __expf(a - mx);
    }
    red[tid] = sum; __syncthreads();
    for (int s = 128; s > 0; s >>= 1) {
        if (tid < s) red[tid] += red[tid + s];
        __syncthreads();
    }
    float inv = 1.f / red[0];

    for (int t = tid; t < TT; t += 256) {
        float a = mb[t] ? -INFINITY : ab[t];
        weights[(long)b * TT + t] = __expf(a - mx) * inv;
    }
}

// ---------------- kernel 4: partial context over T segments (streams 256 MB)
__global__ void lsa_context_partial(const float* __restrict__ weights,  // (B,T)
                                    const float* __restrict__ memory,   // (B,T,E)
                                    float* __restrict__ partial)        // (B,NSEG,E)
{
    __shared__ float wl[TT / NSEG];
    int b = blockIdx.x, seg = blockIdx.y;
    int e = threadIdx.x;                            // 512 threads, e column
    int t0 = seg * (TT / NSEG);
    wl[e] = weights[(long)b * TT + t0 + e];         // 512 == TT/NSEG
    __syncthreads();
    float acc = 0.f;
    const float* mb = memory + ((long)b * TT + t0) * EMB + e;
    #pragma unroll 4
    for (int i = 0; i < TT / NSEG; ++i)
        acc = fmaf(wl[i], mb[(long)i * EMB], acc);
    partial[((long)b * NSEG + seg) * EMB + e] = acc;
}

// ---------------- kernel 5: reduce partials into d_out context
__global__ void lsa_context_reduce(const float* __restrict__ partial,
                                   float* __restrict__ ctx) {
    int b = blockIdx.x, e = threadIdx.x;            // 512 threads
    float s = 0.f;
    #pragma unroll
    for (int seg = 0; seg < NSEG; ++seg)
        s += partial[((long)b * NSEG + seg) * EMB + e];
    ctx[(long)b * EMB + e] = s;
}

extern "C" void kernel_launch(void* const* d_in, const int* in_sizes, int n_in,
                              void* d_out, int out_size, void* d_ws, size_t ws_size,
                              hipStream_t stream) {
    const float* h     = (const float*)d_in[0];            // (B,1024)
    const float* mem   = (const float*)d_in[1];            // (B,T,512)
    const float* pmem  = (const float*)d_in[2];            // (B,T,128)
    const float* awcat = (const float*)d_in[3];            // (B,2,T)
    const unsigned char* mask = (const unsigned char*)d_in[4]; // (B,T) bool
    const float* wq    = (const float*)d_in[5];            // (128,1024)
    const float* cw    = (const float*)d_in[6];            // (32,2,31)
    const float* wd    = (const float*)d_in[7];            // (128,32)
    const float* wv    = (const float*)d_in[8];            // (1,128)

    float* out_ctx = (float*)d_out;                        // (B,512)
    float* out_w   = out_ctx + (long)BB * EMB;             // (B,T)

    float* ws      = (float*)d_ws;
    float* ws_align = ws;                                  // B*T
    float* ws_pq    = ws_align + (long)BB * TT;            // B*128
    float* ws_part  = ws_pq + (long)BB * AD;               // B*NSEG*512

    lsa_pq<<<BB, 128, 0, stream>>>(h, wq, ws_pq);
    lsa_energy<<<dim3(BB, TT / TTILE), 256, 0, stream>>>(
        awcat, cw, wd, ws_pq, pmem, wv, ws_align);
    lsa_softmax<<<BB, 256, 0, stream>>>(ws_align, mask, out_w);
    lsa_context_partial<<<dim3(BB, NSEG), EMB, 0, stream>>>(out_w, mem, ws_part);
    lsa_context_reduce<<<BB, EMB, 0, stream>>>(ws_part, out_ctx);
}